// GCN_42013370090219
// MI455X (gfx1250) — compile-verified
//
#include <hip/hip_runtime.h>
#include <hip/hip_bf16.h>
#include <math.h>

// ---------------------------------------------------------------------------
// GCN forward on MI455X (gfx1250, wave32):
//   h  = relu( A @ (x @ W1) + b1 )
//   z  = A @ (h @ W2) + b2
//   out= log_softmax(z, axis=1)
// Dense GEMMs use V_WMMA_F32_16X16X4_F32; SpMM uses per-edge wave-uniform
// scatter with native global_atomic_add_f32 (gathers hit the 192MB L2).
// NOTE: gcn_scatter1 is deliberately the first kernel in this file so the
// compile-loop disasm snippet shows the atomic codegen.
// ---------------------------------------------------------------------------

typedef float v2f __attribute__((ext_vector_type(2)));
typedef float v8f __attribute__((ext_vector_type(8)));

#define F_IN  512
#define F_HID 256
#define F_OUT 40

// Native, non-returning f32 atomic add: ISA GLOBAL_ATOMIC_ADD_F32 (GV mode,
// 64-bit VGPR address). Non-returning -> tracked by STOREcnt only; S_ENDPGM's
// implicit wait-idle covers completion at kernel end. Inter-kernel ordering is
// provided by kernel launch boundaries on `stream`.
static __device__ __forceinline__ void atomic_fadd_native(float* p, float v) {
    asm volatile("global_atomic_add_f32 %0, %1, off"
                 :: "v"(p), "v"(v)
                 : "memory");
}

// ---------------------------------------------------------------------------
// Scatter 1: agg1[dst] += w * S1[src]   (256 features, one edge per block)
// Edge metadata is block-uniform -> scalar loads; features coalesced
// (one wave = 32 consecutive floats = 128B contiguous gather).
// ---------------------------------------------------------------------------
__global__ __launch_bounds__(256) void gcn_scatter1(
    const int* __restrict__ src, const int* __restrict__ dst,
    const float* __restrict__ ew, const float* __restrict__ S1,
    float* __restrict__ AGG1, int E)
{
    const int e = blockIdx.x;
    if (e >= E) return;
    const int   s = src[e];
    const int   d = dst[e];
    const float w = ew[e];
    const int   c = threadIdx.x;
    const float val = w * S1[(long long)s * F_HID + c];
    atomic_fadd_native(&AGG1[(long long)d * F_HID + c], val);
}

// ---------------------------------------------------------------------------
// fp32 WMMA 16x16x4:  D(16x16) = A(16x4) * B(4x16) + C
// A layout (per ISA 7.12.2, 32-bit A 16x4): lane = 16*kHi + m,
//   vgpr0 -> K = 2*kHi, vgpr1 -> K = 2*kHi+1   (kHi = lane>>4, m = lane&15)
// B layout (row striped across lanes): lane = 16*kHi + n,
//   vgpr0 -> K = 2*kHi, vgpr1 -> K = 2*kHi+1
// C/D layout: vgpr v -> row v + 8*(lane>>4), col = lane&15
// ---------------------------------------------------------------------------
static __device__ __forceinline__ v8f wmma_f32(v2f a, v2f b, v8f c) {
    return __builtin_amdgcn_wmma_f32_16x16x4_f32(
        /*neg_a=*/false, a, /*neg_b=*/false, b,
        /*c_mod=*/(short)0, c, /*reuse_a=*/false, /*reuse_b=*/false);
}

// ---------------------------------------------------------------------------
// GEMM1: S1[M, 256] = X[M, 512] @ W1[512, 256]
// Block = 256 threads = 8 waves. Each wave: 16 rows x 64 cols (4 N-tiles).
// Grid: (ceil(M/128), 4)
// ---------------------------------------------------------------------------
__global__ __launch_bounds__(256) void gcn_gemm1(
    const float* __restrict__ X, const float* __restrict__ W1,
    float* __restrict__ S1, int M)
{
    const int wave   = threadIdx.x >> 5;
    const int lane   = threadIdx.x & 31;
    const int laneLo = lane & 15;
    const int laneHi = lane >> 4;           // 0 or 1

    const int m0 = blockIdx.x * 128 + wave * 16;
    const int n0 = blockIdx.y * 64;

    int row  = m0 + laneLo;
    int arow = (row < M) ? row : (M - 1);   // clamp loads; stores are guarded

    v8f acc0 = {}, acc1 = {}, acc2 = {}, acc3 = {};

    const float* xrow = X + (long long)arow * F_IN;

    for (int k = 0; k < F_IN; k += 4) {
        const int kk = k + laneHi * 2;
        v2f a;
        a.x = xrow[kk];
        a.y = xrow[kk + 1];

        const float* wk  = W1 + (long long)kk * F_HID + n0 + laneLo;
        v2f b0, b1v, b2v, b3;
        b0.x  = wk[0];           b0.y  = wk[F_HID];
        b1v.x = wk[16];          b1v.y = wk[F_HID + 16];
        b2v.x = wk[32];          b2v.y = wk[F_HID + 32];
        b3.x  = wk[48];          b3.y  = wk[F_HID + 48];

        acc0 = wmma_f32(a, b0,  acc0);
        acc1 = wmma_f32(a, b1v, acc1);
        acc2 = wmma_f32(a, b2v, acc2);
        acc3 = wmma_f32(a, b3,  acc3);
    }

    #pragma unroll
    for (int v = 0; v < 8; ++v) {
        const int r = m0 + v + 8 * laneHi;
        if (r < M) {
            float* o = S1 + (long long)r * F_HID + n0 + laneLo;
            o[0]  = acc0[v];
            o[16] = acc1[v];
            o[32] = acc2[v];
            o[48] = acc3[v];
        }
    }
}

// ---------------------------------------------------------------------------
// GEMM2: S2[M, 40] = relu(AGG1 + b1)[M, 256] @ W2[256, 40]
// relu+bias fused into the A-fragment load. Per wave: 16 rows x 48 cols
// (3 N-tiles, cols >= 40 are guarded/zeroed). Output stride = 40.
// ---------------------------------------------------------------------------
__global__ __launch_bounds__(256) void gcn_gemm2(
    const float* __restrict__ AGG1, const float* __restrict__ b1,
    const float* __restrict__ W2, float* __restrict__ S2, int M)
{
    const int wave   = threadIdx.x >> 5;
    const int lane   = threadIdx.x & 31;
    const int laneLo = lane & 15;
    const int laneHi = lane >> 4;

    const int m0 = blockIdx.x * 128 + wave * 16;

    int row  = m0 + laneLo;
    int arow = (row < M) ? row : (M - 1);

    v8f acc0 = {}, acc1 = {}, acc2 = {};

    const float* hrow = AGG1 + (long long)arow * F_HID;

    for (int k = 0; k < F_HID; k += 4) {
        const int kk = k + laneHi * 2;
        v2f a;
        a.x = fmaxf(hrow[kk]     + b1[kk],     0.0f);
        a.y = fmaxf(hrow[kk + 1] + b1[kk + 1], 0.0f);

        v2f b0 = {}, b1v = {}, b2v = {};
        {
            const int c0 = laneLo;          // tile 0: cols 0..15  (always valid)
            b0.x = W2[(long long)kk * F_OUT + c0];
            b0.y = W2[(long long)(kk + 1) * F_OUT + c0];
            const int c1 = 16 + laneLo;     // tile 1: cols 16..31 (always valid)
            b1v.x = W2[(long long)kk * F_OUT + c1];
            b1v.y = W2[(long long)(kk + 1) * F_OUT + c1];
            const int c2 = 32 + laneLo;     // tile 2: cols 32..47 (guard >= 40)
            if (c2 < F_OUT) {
                b2v.x = W2[(long long)kk * F_OUT + c2];
                b2v.y = W2[(long long)(kk + 1) * F_OUT + c2];
            }
        }

        acc0 = wmma_f32(a, b0,  acc0);
        acc1 = wmma_f32(a, b1v, acc1);
        acc2 = wmma_f32(a, b2v, acc2);
    }

    #pragma unroll
    for (int v = 0; v < 8; ++v) {
        const int r = m0 + v + 8 * laneHi;
        if (r < M) {
            float* o = S2 + (long long)r * F_OUT;
            o[laneLo]      = acc0[v];
            o[16 + laneLo] = acc1[v];
            if (32 + laneLo < F_OUT) o[32 + laneLo] = acc2[v];
        }
    }
}

// ---------------------------------------------------------------------------
// Scatter 2: agg2[dst] += w * S2[src]  (40 features; 4 edges per 256-thr block,
// 64 lanes per edge so src/dst stay wave-uniform)
// ---------------------------------------------------------------------------
__global__ __launch_bounds__(256) void gcn_scatter2(
    const int* __restrict__ src, const int* __restrict__ dst,
    const float* __restrict__ ew, const float* __restrict__ S2,
    float* __restrict__ AGG2, int E)
{
    const int e = blockIdx.x * 4 + (threadIdx.x >> 6);
    const int c = threadIdx.x & 63;
    if (e >= E || c >= F_OUT) return;
    const int   s = src[e];
    const int   d = dst[e];
    const float w = ew[e];
    const float val = w * S2[(long long)s * F_OUT + c];
    atomic_fadd_native(&AGG2[(long long)d * F_OUT + c], val);
}

// ---------------------------------------------------------------------------
// out = log_softmax(AGG2 + b2, axis=1): one wave32 per row (40 logits ->
// lanes handle col and col+32), __shfl_xor tree reductions.
// ---------------------------------------------------------------------------
__global__ __launch_bounds__(256) void gcn_logsoftmax(
    const float* __restrict__ AGG2, const float* __restrict__ b2,
    float* __restrict__ out, int M)
{
    const int wave = threadIdx.x >> 5;
    const int lane = threadIdx.x & 31;
    const int row  = blockIdx.x * 8 + wave;
    if (row >= M) return;

    const float* z = AGG2 + (long long)row * F_OUT;
    const bool hi_ok = (lane + 32) < F_OUT;

    float v0 = z[lane] + b2[lane];
    float v1 = hi_ok ? (z[lane + 32] + b2[lane + 32]) : -3.402823466e38f;

    float m = fmaxf(v0, v1);
    #pragma unroll
    for (int off = 16; off > 0; off >>= 1)
        m = fmaxf(m, __shfl_xor(m, off, 32));

    float s = expf(v0 - m) + (hi_ok ? expf(v1 - m) : 0.0f);
    #pragma unroll
    for (int off = 16; off > 0; off >>= 1)
        s += __shfl_xor(s, off, 32);

    const float ls = logf(s);
    float* o = out + (long long)row * F_OUT;
    o[lane] = v0 - m - ls;
    if (hi_ok) o[lane + 32] = v1 - m - ls;
}

// ---------------------------------------------------------------------------
// Host-side orchestration (graph-capture safe: only kernel launches and
// hipMemsetAsync on `stream`).
// ---------------------------------------------------------------------------
extern "C" void kernel_launch(void* const* d_in, const int* in_sizes, int n_in,
                              void* d_out, int out_size, void* d_ws, size_t ws_size,
                              hipStream_t stream) {
    const float* x    = (const float*)d_in[0];   // [M, 512]
    const int*   ei   = (const int*)  d_in[1];   // [2, E]
    const float* ew   = (const float*)d_in[2];   // [E]
    const float* W1   = (const float*)d_in[3];   // [512, 256]
    const float* b1   = (const float*)d_in[4];   // [256]
    const float* W2   = (const float*)d_in[5];   // [256, 40]
    const float* b2   = (const float*)d_in[6];   // [40]
    float*       out  = (float*)d_out;           // [M, 40]

    const int M = in_sizes[0] / F_IN;            // 50000
    const int E = in_sizes[2];                   // 800000
    const int* src = ei;
    const int* dst = ei + E;

    // Workspace: bufA = support (S1 then S2), bufB = aggregate (agg1 then agg2)
    char*  ws   = (char*)d_ws;
    size_t bufN = (size_t)M * F_HID * sizeof(float);   // 51.2 MB each
    float* bufA = (float*)ws;
    float* bufB = (float*)(ws + bufN);

    const int mblocks = (M + 127) / 128;

    // 1) S1 = x @ W1
    gcn_gemm1<<<dim3(mblocks, F_HID / 64), 256, 0, stream>>>(x, W1, bufA, M);

    // 2) agg1 = A @ S1
    hipMemsetAsync(bufB, 0, (size_t)M * F_HID * sizeof(float), stream);
    gcn_scatter1<<<E, 256, 0, stream>>>(src, dst, ew, bufA, bufB, E);

    // 3) S2 = relu(agg1 + b1) @ W2   (S2 reuses bufA; S1 is dead)
    gcn_gemm2<<<mblocks, 256, 0, stream>>>(bufB, b1, W2, bufA, M);

    // 4) agg2 = A @ S2   (agg2 reuses bufB; agg1 is dead after gemm2)
    hipMemsetAsync(bufB, 0, (size_t)M * F_OUT * sizeof(float), stream);
    gcn_scatter2<<<(E + 3) / 4, 256, 0, stream>>>(src, dst, ew, bufA, bufB, E);

    // 5) out = log_softmax(agg2 + b2)
    gcn_logsoftmax<<<(M + 7) / 8, 256, 0, stream>>>(bufB, b2, out, M);
}